// Layered_MS_Decoder_42606075576371
// MI455X (gfx1250) — compile-verified
//
#include <hip/hip_runtime.h>
#include <stdint.h>

// Problem constants (from reference): M=512, N=1024, W=8, B=2048, ITER=5
#define M_CN    512
#define N_VN    1024
#define W_DEG   8
#define B_BATCH 2048
#define N_ITER  5

// Mapping: one wave32 workgroup = 4 batch elements x 8 edges
#define BPW       4
#define NTHREADS  32
#define C2V_STRIDE (M_CN * W_DEG + 16)          // 4112 dwords; 4112 % 64 == 16 -> bank-staggered rows

// LDS layout (dwords)
#define OFF_VN   0                               // vn: n-major, vn[n*BPW + b]      (4096)
#define OFF_C2V  (N_VN * BPW)                    // c2v: per-b padded rows          (4*4112)
#define OFF_H    (OFF_C2V + BPW * C2V_STRIDE)    // H_compact int32                 (4096)
#define OFF_CN   (OFF_H + M_CN * W_DEG)          // cn_order int32                  (512)
#define LDS_DWORDS (OFF_CN + M_CN)               // 25152 dwords
#define SMEM_BYTES (LDS_DWORDS * 4)              // 100608 B  (<< 320KB per WGP)

typedef unsigned int u32;
typedef unsigned long long u64;
typedef u32 v4u __attribute__((ext_vector_type(4)));
typedef int  v8i __attribute__((ext_vector_type(8)));
typedef int  v4i __attribute__((ext_vector_type(4)));

#if defined(__HIP_DEVICE_COMPILE__) && __has_builtin(__builtin_amdgcn_tensor_load_to_lds)
#define HAVE_TDM 1
#else
#define HAVE_TDM 0
#endif

#if HAVE_TDM
// 1-D TDM copy: nelem elements of (1<<dsz_code) bytes from global -> LDS.
// D# built per CDNA5 ISA (group0: count/lds/global/type,
// group1: data_size, tensor_dim0, tensor_dim1=1, tile_dim0=nelem, tile_dim1=0 (1D)).
// 6-arg builtin on this toolchain: (u32x4 g0, i32x8 g1, i32x4, i32x4, i32x8, i32 cpol)
__device__ __forceinline__ void tdm_load_1d(u32 lds_byte, u64 gaddr, u32 nelem, u32 dsz_code)
{
    v4u g0;
    g0.x = 1u;                                               // count=1, is_restore=0
    g0.y = lds_byte;                                         // lds_addr
    g0.z = (u32)(gaddr & 0xffffffffull);                     // global_addr[31:0]
    g0.w = (u32)((gaddr >> 32) & 0x1ffffffull) | (2u << 30); // global_addr[56:32] | type=2
    v8i g1;
    g1[0] = (int)(dsz_code << 16);                           // workgroup_mask=0 (not in cluster), data_size
    g1[1] = (int)((nelem & 0xffffu) << 16);                  // tensor_dim0[15:0]
    g1[2] = (int)((nelem >> 16) | (1u << 16));               // tensor_dim0[31:16] | tensor_dim1=1
    g1[3] = (int)((nelem & 0xffffu) << 16);                  // tensor_dim1 hi=0 | tile_dim0=nelem
    g1[4] = 0;                                               // tile_dim1=0 (1D), tile_dim2=0
    g1[5] = (int)nelem;                                      // tensor_dim0_stride lo
    g1[6] = 0;                                               // stride0 hi | stride1 lo
    g1[7] = 0;                                               // stride1 hi
    v4i z4 = {0, 0, 0, 0};
    v8i z8 = {0, 0, 0, 0, 0, 0, 0, 0};
    __builtin_amdgcn_tensor_load_to_lds(g0, g1, z4, z4, z8, 0);
}
#endif

__global__ __launch_bounds__(NTHREADS) void ldpc_layered_ms(
    const float* __restrict__ llr,      // (B, N)
    const float* __restrict__ alphas,   // (ITER,)
    const int*   __restrict__ cn_order, // (M,)   int32 (JAX default x32)
    const int*   __restrict__ Hc,       // (M, W) int32
    float*       __restrict__ out)      // (ITER, B, N)
{
    extern __shared__ float smem[];
    float* vn  = smem + OFF_VN;
    float* c2v = smem + OFF_C2V;
    int*   Hl  = (int*)(smem + OFF_H);
    int*   cnl = (int*)(smem + OFF_CN);

    const int tid = threadIdx.x;
    const int bl  = tid >> 3;    // batch slot within WG: 0..3
    const int w   = tid & 7;     // edge index:          0..7
    const int wgbase = blockIdx.x * BPW;

    __builtin_prefetch(llr + ((size_t)wgbase << 10), 0, 1);  // -> global_prefetch_b8

#if HAVE_TDM
    // DMA the shared tables into LDS via the Tensor Data Mover (both in flight,
    // one TENSORcnt wait covers both; same-wave tensor ops retire in order).
    const u32 lds_base = __builtin_amdgcn_groupstaticsize();
    tdm_load_1d(lds_base + (u32)(OFF_CN * 4), (u64)(uintptr_t)cn_order, (u32)M_CN, 2u);
    tdm_load_1d(lds_base + (u32)(OFF_H  * 4), (u64)(uintptr_t)Hc, (u32)(M_CN * W_DEG), 2u);
#else
    for (int i = tid; i < M_CN; i += NTHREADS) cnl[i] = cn_order[i];
    for (int i = tid; i < M_CN * W_DEG; i += NTHREADS) Hl[i] = Hc[i];
#endif

    // Init vn from channel LLRs: b128 global reads (512B/wave/instr), 4 dword
    // LDS scatters into the n-major layout.
    {
        const float4* llr4 = (const float4*)(llr + ((size_t)wgbase << 10));
        for (int i = tid; i < (BPW * N_VN) / 4; i += NTHREADS) {
            int b = i >> 8, n0 = (i & 255) << 2;
            float4 v = llr4[(b << 8) | (i & 255)];
            vn[(n0 + 0) * BPW + b] = v.x;
            vn[(n0 + 1) * BPW + b] = v.y;
            vn[(n0 + 2) * BPW + b] = v.z;
            vn[(n0 + 3) * BPW + b] = v.w;
        }
    }
    // Zero c2v state (incl. pad).
    for (int i = tid; i < BPW * C2V_STRIDE; i += NTHREADS) c2v[i] = 0.0f;

#if HAVE_TDM
    __builtin_amdgcn_s_wait_tensorcnt(0);
#endif
    __builtin_amdgcn_wave_barrier();   // compiler fence; HW keeps same-wave DS ops in order

    for (int it = 0; it < N_ITER; ++it) {
        const float as = 1.0f / (1.0f + expf(-alphas[it]));   // sigmoid(alpha)

        for (int k = 0; k < M_CN; ++k) {
            const int   cn  = cnl[k];
            const int   idx = Hl[(cn << 3) + w];
            const int   cvo = bl * C2V_STRIDE + (cn << 3) + w;
            const float old = c2v[cvo];
            const float cur = vn[idx * BPW + bl] - old;

            const float av = fabsf(cur);
            const u32   sb = __float_as_uint(cur) >> 31;

            // Butterfly (min1, min2, sign-xor) over the 8 edge lanes of this batch slot.
            float m1 = av, m2 = __builtin_inff();
            u32   sx = sb;
#pragma unroll
            for (int off = 1; off <= 4; off <<= 1) {
                const float om1 = __shfl_xor(m1, off, 32);
                const float om2 = __shfl_xor(m2, off, 32);
                const u32   osx = (u32)__shfl_xor((int)sx, off, 32);
                const float hi  = fmaxf(m1, om1);
                m1 = fminf(m1, om1);
                m2 = fminf(fminf(m2, om2), hi);
                sx ^= osx;
            }

            // Exclude-self: if this lane holds the min, use second-min (exact on ties).
            const float amp = (av == m1) ? m2 : m1;
            const u32   so  = (sx ^ sb) & 1u;              // sign product of the other 7
            float c = fminf(amp * as, 20.0f);              // clip(|.|*a, 20), amp>=0
            c = __uint_as_float(__float_as_uint(c) | (so << 31));

            vn[idx * BPW + bl] = cur + c;
            c2v[cvo] = c;
            __builtin_amdgcn_wave_barrier();               // order step k stores vs k+1 loads
        }

        // Dump vn for this iteration: gather 4 n-major dwords, one b128 store
        // per lane (contiguous 512B per wave per instruction).
        float4* orow4 = (float4*)(out + (size_t)it * ((size_t)B_BATCH * N_VN)
                                      + ((size_t)wgbase << 10));
        for (int i = tid; i < (BPW * N_VN) / 4; i += NTHREADS) {
            int b = i >> 8, n0 = (i & 255) << 2;
            float4 v;
            v.x = vn[(n0 + 0) * BPW + b];
            v.y = vn[(n0 + 1) * BPW + b];
            v.z = vn[(n0 + 2) * BPW + b];
            v.w = vn[(n0 + 3) * BPW + b];
            orow4[(b << 8) | (i & 255)] = v;
        }
        __builtin_amdgcn_wave_barrier();
    }
}

extern "C" void kernel_launch(void* const* d_in, const int* in_sizes, int n_in,
                              void* d_out, int out_size, void* d_ws, size_t ws_size,
                              hipStream_t stream)
{
    (void)in_sizes; (void)n_in; (void)d_ws; (void)ws_size; (void)out_size;
    const float* llr    = (const float*)d_in[0];
    const float* alphas = (const float*)d_in[1];
    const int*   cnord  = (const int*)d_in[2];
    const int*   Hc     = (const int*)d_in[3];
    // d_in[4] (mask) is implied by the exclude-self reduction; unused.
    float* out = (float*)d_out;

    (void)hipFuncSetAttribute(reinterpret_cast<const void*>(ldpc_layered_ms),
                              hipFuncAttributeMaxDynamicSharedMemorySize, SMEM_BYTES);

    ldpc_layered_ms<<<B_BATCH / BPW, NTHREADS, SMEM_BYTES, stream>>>(
        llr, alphas, cnord, Hc, out);
}